// TwoLayerDirectedGCN_52201032515657
// MI455X (gfx1250) — compile-verified
//
#include <hip/hip_runtime.h>

typedef float v2f __attribute__((ext_vector_type(2)));
typedef float v8f __attribute__((ext_vector_type(8)));

#define DFEAT 128

// ---------------------------------------------------------------------------
// deg[row[e]] += w[e]
__global__ __launch_bounds__(256) void deg_accum_kernel(const int* __restrict__ row,
                                                        const float* __restrict__ w,
                                                        float* __restrict__ deg, int E) {
    int e = blockIdx.x * 256 + threadIdx.x;
    if (e < E) atomicAdd(&deg[row[e]], w[e]);
}

// deg[i] = deg[i]^-0.5  (deg > 0 guaranteed by construction)
__global__ __launch_bounds__(256) void rsqrt_kernel(float* __restrict__ deg, int n) {
    int i = blockIdx.x * 256 + threadIdx.x;
    if (i < n) deg[i] = 1.0f / sqrtf(deg[i]);
}

// norm[e] = dinv[row[e]] * w[e] * dinv[col[e]]
__global__ __launch_bounds__(256) void norm_kernel(const int* __restrict__ row,
                                                   const int* __restrict__ col,
                                                   const float* __restrict__ w,
                                                   const float* __restrict__ dinv,
                                                   float* __restrict__ norm, int E) {
    int e = blockIdx.x * 256 + threadIdx.x;
    if (e < E) norm[e] = dinv[row[e]] * w[e] * dinv[col[e]];
}

// ---------------------------------------------------------------------------
// H = (RELU ? max(X,0) : X) @ W^T + b   with V_WMMA_F32_16X16X4_F32
// Block: 256 threads = 8 waves. Block owns 16 rows; wave w owns columns
// [16w, 16w+16). K = 128, stepped by 4 (32 WMMAs per tile).
//
// A 16x4 f32 fragment (2 VGPRs/lane): lane holds (m = lane%16,
//   k = kb + 2*(lane/16) + {0,1})  -> contiguous float2 from x row m.
// B 4x16 f32 fragment (2 VGPRs/lane): lane holds (n = lane%16,
//   k = kb + 2*(lane/16) + {0,1})  -> contiguous float2 from W row col0+n,
//   since B[k][n] = W[col0+n][k] (NT gemm; W rows already column-major B).
// C/D 16x16 f32: vgpr r, lane L -> (M = r + 8*(L/16), N = L%16).
template <bool RELU>
__global__ __launch_bounds__(256) void gemm_wmma_kernel(const float* __restrict__ X,
                                                        const float* __restrict__ W,
                                                        const float* __restrict__ bias,
                                                        float* __restrict__ H, int nrows) {
    const int lane = threadIdx.x & 31;
    const int wave = threadIdx.x >> 5;
    const int m    = lane & 15;
    const int half = lane >> 4;          // 0 or 1
    const int row0 = blockIdx.x * 16;
    const int col0 = wave * 16;

    int ar = row0 + m;
    if (ar >= nrows) ar = nrows - 1;     // clamp loads; stores guarded below
    const float* __restrict__ xrow = X + (size_t)ar * DFEAT;
    const float* __restrict__ wrow = W + (size_t)(col0 + m) * DFEAT;

    const float bv = bias[col0 + m];
    v8f c;
#pragma unroll
    for (int r = 0; r < 8; ++r) c[r] = bv;

#pragma unroll
    for (int kb = 0; kb < DFEAT; kb += 4) {
        v2f a = *(const v2f*)(xrow + kb + 2 * half);
        if (RELU) { a.x = fmaxf(a.x, 0.0f); a.y = fmaxf(a.y, 0.0f); }
        v2f b = *(const v2f*)(wrow + kb + 2 * half);
        // (neg_a, A, neg_b, B, c_mod, C, reuse_a, reuse_b)
        c = __builtin_amdgcn_wmma_f32_16x16x4_f32(false, a, false, b,
                                                  (short)0, c, false, false);
    }

#pragma unroll
    for (int r = 0; r < 8; ++r) {
        int M = row0 + r + 8 * half;
        if (M < nrows) H[(size_t)M * DFEAT + col0 + m] = c[r];
    }
}

// ---------------------------------------------------------------------------
// out[col[e]][:] += norm[e] * H[row[e]][:]
// One wave per edge; lane handles 4 contiguous features (float4 gather,
// 4x global_atomic_add_f32 scatter). All targets L2-resident (25.6 MB < 192 MB).
__global__ __launch_bounds__(256) void scatter_add_kernel(const float* __restrict__ H,
                                                          const int* __restrict__ row,
                                                          const int* __restrict__ col,
                                                          const float* __restrict__ norm,
                                                          float* __restrict__ out, int E) {
    int e = blockIdx.x * 8 + (threadIdx.x >> 5);
    if (e >= E) return;
    const int lane = threadIdx.x & 31;
    const int r = row[e];
    const int cidx = col[e];
    const float nv = norm[e];

    const float4 hv = *(const float4*)(H + (size_t)r * DFEAT + lane * 4);
    float* op = out + (size_t)cidx * DFEAT + lane * 4;
    atomicAdd(op + 0, nv * hv.x);
    atomicAdd(op + 1, nv * hv.y);
    atomicAdd(op + 2, nv * hv.z);
    atomicAdd(op + 3, nv * hv.w);
}

// ---------------------------------------------------------------------------
extern "C" void kernel_launch(void* const* d_in, const int* in_sizes, int n_in,
                              void* d_out, int out_size, void* d_ws, size_t ws_size,
                              hipStream_t stream) {
    const float* x  = (const float*)d_in[0];   // [N,128]
    const int*   ei = (const int*)d_in[1];     // [2,E]
    const float* ew = (const float*)d_in[2];   // [E]
    const float* W1 = (const float*)d_in[3];   // [128,128]
    const float* b1 = (const float*)d_in[4];   // [128]
    const float* W2 = (const float*)d_in[5];   // [128,128]
    const float* b2 = (const float*)d_in[6];   // [128]
    float* out = (float*)d_out;

    const int E = in_sizes[2];
    const int n = in_sizes[0] / DFEAT;
    const int* row = ei;
    const int* col = ei + E;

    // Workspace carve-up (256B aligned): deg[n], norm[E], h[n*128], agg[n*128]
    char* ws = (char*)d_ws;
    size_t off = 0;
    auto alignup = [](size_t v) { return (v + 255) & ~(size_t)255; };
    float* deg  = (float*)(ws + off); off += alignup((size_t)n * 4);
    float* norm = (float*)(ws + off); off += alignup((size_t)E * 4);
    float* h    = (float*)(ws + off); off += alignup((size_t)n * DFEAT * 4);
    float* agg  = (float*)(ws + off);

    const int gemm_blocks = (n + 15) / 16;
    const int edge_blocks = (E + 255) / 256;
    const int node_blocks = (n + 255) / 256;
    const int scat_blocks = (E + 7) / 8;

    // --- normalization coefficients ---
    hipMemsetAsync(deg, 0, (size_t)n * 4, stream);
    deg_accum_kernel<<<edge_blocks, 256, 0, stream>>>(row, ew, deg, E);
    rsqrt_kernel<<<node_blocks, 256, 0, stream>>>(deg, n);
    norm_kernel<<<edge_blocks, 256, 0, stream>>>(row, col, ew, deg, norm, E);

    // --- layer 1: GEMM + aggregate ---
    gemm_wmma_kernel<false><<<gemm_blocks, 256, 0, stream>>>(x, W1, b1, h, n);
    hipMemsetAsync(agg, 0, (size_t)n * DFEAT * 4, stream);
    scatter_add_kernel<<<scat_blocks, 256, 0, stream>>>(h, row, col, norm, agg, E);

    // --- layer 2: ReLU fused into GEMM A-load, then aggregate into d_out ---
    gemm_wmma_kernel<true><<<gemm_blocks, 256, 0, stream>>>(agg, W2, b2, h, n);
    hipMemsetAsync(out, 0, (size_t)n * DFEAT * 4, stream);
    scatter_add_kernel<<<scat_blocks, 256, 0, stream>>>(h, row, col, norm, out, E);
}